// GAT_66769561583833
// MI455X (gfx1250) — compile-verified
//
#include <hip/hip_runtime.h>
#include <math.h>

// Problem constants (match reference)
#define N_NODES   50000
#define N_EDGES   800000
#define F_IN      128
#define H_DIM     128
#define DE_DIM    16
#define C_CLS     2
#define G_GRAPHS  32
#define NEG_SLOPE 0.2f
#define ROWS_TOT  (N_EDGES + N_NODES)   // edges + self loops

typedef __attribute__((ext_vector_type(2))) float v2f;
typedef __attribute__((ext_vector_type(8))) float v8f;

// ---------------------------------------------------------------------------
// Float atomic-max via int ordering trick (works for mixed signs, -inf init)
// ---------------------------------------------------------------------------
__device__ __forceinline__ void atomicMaxF(float* addr, float val) {
    if (val >= 0.0f) atomicMax((int*)addr, __float_as_int(val));
    else             atomicMin((unsigned int*)addr, __float_as_uint(val));
}

// ---------------------------------------------------------------------------
// Generic fill
// ---------------------------------------------------------------------------
__global__ void fill_kernel(float* __restrict__ p, long long n, float v) {
    long long i = (long long)blockIdx.x * blockDim.x + threadIdx.x;
    if (i < n) p[i] = v;
}

// ---------------------------------------------------------------------------
// K1: degree + scatter-sum of edge_attr over dst (for self-loop fill='mean')
// one thread per (edge, attr-dim)
// ---------------------------------------------------------------------------
__global__ void edge_deg_kernel(const int* __restrict__ dstp,
                                const float* __restrict__ ea,
                                float* __restrict__ deg,
                                float* __restrict__ loop) {
    int idx = blockIdx.x * blockDim.x + threadIdx.x;       // E * DE
    if (idx >= N_EDGES * DE_DIM) return;
    int e = idx >> 4, d = idx & (DE_DIM - 1);
    int t = dstp[e];
    atomicAdd(&loop[t * DE_DIM + d], ea[idx]);
    if (d == 0) atomicAdd(&deg[t], 1.0f);
}

// K2: loop_attr = loop_sum / max(deg,1)   (in place)
__global__ void loop_div_kernel(float* __restrict__ loop,
                                const float* __restrict__ deg) {
    int idx = blockIdx.x * blockDim.x + threadIdx.x;
    if (idx >= N_NODES * DE_DIM) return;
    loop[idx] /= fmaxf(deg[idx >> 4], 1.0f);
}

// ---------------------------------------------------------------------------
// K3: xl = x@Wl+bl, xr = x@Wr+br  via V_WMMA_F32_16X16X4_F32
// grid = N/16 blocks, 256 threads = 8 waves; wave w owns n-tile w (H=128).
// A 16x4 layout: lane<16 -> (K=k,k+1) of row m0+lane ; lane>=16 -> (K=k+2,k+3)
// B 4x16 layout mirrored: v0/v1 = rows (k+2*half, k+2*half+1), col n0+(lane&15)
// D: VGPR v -> row m0 + v + 8*half, col n0 + (lane&15)
// EXEC all ones: grid exactly covers 50000 rows, no divergence in this kernel.
// ---------------------------------------------------------------------------
__global__ __launch_bounds__(256) void gemm_xlxr_kernel(
    const float* __restrict__ x,
    const float* __restrict__ Wl, const float* __restrict__ bl,
    const float* __restrict__ Wr, const float* __restrict__ br,
    float* __restrict__ xl, float* __restrict__ xr)
{
    const int lane = threadIdx.x & 31;
    const int wave = threadIdx.x >> 5;
    const int m0   = blockIdx.x * 16;
    const int n0   = wave * 16;
    const int half = lane >> 4;
    const int l    = lane & 15;

    v8f accL = {};
    v8f accR = {};
    const float* xrow = x + (size_t)(m0 + l) * F_IN + 2 * half;

    #pragma unroll
    for (int k = 0; k < F_IN; k += 4) {
        v2f a = *(const v2f*)(xrow + k);                    // 8B-aligned
        int kb = k + 2 * half;
        v2f bLv = { Wl[(size_t)kb * H_DIM + n0 + l],
                    Wl[(size_t)(kb + 1) * H_DIM + n0 + l] };
        v2f bRv = { Wr[(size_t)kb * H_DIM + n0 + l],
                    Wr[(size_t)(kb + 1) * H_DIM + n0 + l] };
        accL = __builtin_amdgcn_wmma_f32_16x16x4_f32(
            false, a, false, bLv, (short)0, accL, false, false);
        accR = __builtin_amdgcn_wmma_f32_16x16x4_f32(
            false, a, false, bRv, (short)0, accR, false, false);
    }

    const int n = n0 + l;
    const float bLb = bl[n], bRb = br[n];
    #pragma unroll
    for (int v = 0; v < 8; v++) {
        int m = m0 + v + 8 * half;
        xl[(size_t)m * H_DIM + n] = accL[v] + bLb;
        xr[(size_t)m * H_DIM + n] = accR[v] + bRb;
    }
}

// ---------------------------------------------------------------------------
// K4: per-edge logit = leaky(xl[src]+xr[dst]+ea@We) . att ; segment atomic max
// one wave per edge row; We (16x128) + att staged in LDS
// ---------------------------------------------------------------------------
__global__ __launch_bounds__(256) void edge_logit_kernel(
    const int* __restrict__ srcp, const int* __restrict__ dstp,
    const float* __restrict__ ea, const float* __restrict__ loop,
    const float* __restrict__ xl, const float* __restrict__ xr,
    const float* __restrict__ We, const float* __restrict__ att,
    float* __restrict__ lg, float* __restrict__ mx)
{
    __shared__ __align__(16) float sWe[DE_DIM * H_DIM];
    __shared__ __align__(16) float sAtt[H_DIM];
    for (int i = threadIdx.x; i < DE_DIM * H_DIM; i += blockDim.x) sWe[i] = We[i];
    if (threadIdx.x < H_DIM) sAtt[threadIdx.x] = att[threadIdx.x];
    __syncthreads();

    const int lane = threadIdx.x & 31;
    const int wave = threadIdx.x >> 5;
    const long long r = (long long)blockIdx.x * 8 + wave;
    if (r >= ROWS_TOT) return;

    int s, t; const float* earow;
    if (r < N_EDGES) { s = srcp[r]; t = dstp[r]; earow = ea + r * DE_DIM; }
    else             { s = (int)(r - N_EDGES); t = s; earow = loop + (size_t)s * DE_DIM; }

    float eav = (lane < DE_DIM) ? earow[lane] : 0.0f;

    const int h0 = lane * 4;
    float4 vl = *(const float4*)(xl + (size_t)s * H_DIM + h0);
    float4 vr = *(const float4*)(xr + (size_t)t * H_DIM + h0);
    float m0v = vl.x + vr.x, m1v = vl.y + vr.y;
    float m2v = vl.z + vr.z, m3v = vl.w + vr.w;

    #pragma unroll
    for (int k = 0; k < DE_DIM; k++) {
        float ek = __shfl(eav, k, 32);
        float4 w = *(const float4*)(sWe + k * H_DIM + h0);
        m0v += ek * w.x; m1v += ek * w.y; m2v += ek * w.z; m3v += ek * w.w;
    }

    float4 aw = *(const float4*)(sAtt + h0);
    float sacc = (m0v > 0.f ? m0v : NEG_SLOPE * m0v) * aw.x
               + (m1v > 0.f ? m1v : NEG_SLOPE * m1v) * aw.y
               + (m2v > 0.f ? m2v : NEG_SLOPE * m2v) * aw.z
               + (m3v > 0.f ? m3v : NEG_SLOPE * m3v) * aw.w;

    #pragma unroll
    for (int off = 16; off > 0; off >>= 1)
        sacc += __shfl_xor(sacc, off, 32);

    if (lane == 0) {
        lg[r] = sacc;
        atomicMaxF(&mx[t], sacc);
    }
}

// K5: ex = exp(logit - mx[dst]) (in place), den[dst] += ex
__global__ void edge_exp_kernel(const int* __restrict__ dstp,
                                float* __restrict__ lg,
                                const float* __restrict__ mx,
                                float* __restrict__ den) {
    int r = blockIdx.x * blockDim.x + threadIdx.x;
    if (r >= ROWS_TOT) return;
    int t = (r < N_EDGES) ? dstp[r] : (r - N_EDGES);
    float ex = __expf(lg[r] - mx[t]);
    lg[r] = ex;
    atomicAdd(&den[t], ex);
}

// K6: out[dst] += (ex/den[dst]) * xl[src] ; one thread per (edge,h)
__global__ void edge_agg_kernel(const int* __restrict__ srcp,
                                const int* __restrict__ dstp,
                                const float* __restrict__ lg,
                                const float* __restrict__ den,
                                const float* __restrict__ xl,
                                float* __restrict__ outacc) {
    long long idx = (long long)blockIdx.x * blockDim.x + threadIdx.x;
    if (idx >= (long long)ROWS_TOT * H_DIM) return;
    int r = (int)(idx >> 7);
    int h = (int)(idx & (H_DIM - 1));
    int s, t;
    if (r < N_EDGES) { s = srcp[r]; t = dstp[r]; }
    else             { s = r - N_EDGES; t = s; }
    float alpha = lg[r] / den[t];
    atomicAdd(&outacc[(size_t)t * H_DIM + h], alpha * xl[(size_t)s * H_DIM + h]);
}

// K7: relu(out + bias_conv) -> per-graph sum + counts
__global__ void node_pool_kernel(const float* __restrict__ outacc,
                                 const float* __restrict__ bias,
                                 const int* __restrict__ batch,
                                 float* __restrict__ pooled,
                                 float* __restrict__ cnt) {
    int idx = blockIdx.x * blockDim.x + threadIdx.x;
    if (idx >= N_NODES * H_DIM) return;
    int n = idx >> 7, h = idx & (H_DIM - 1);
    float v = fmaxf(outacc[idx] + bias[h], 0.0f);
    int g = batch[n];
    atomicAdd(&pooled[g * H_DIM + h], v);
    if (h == 0) atomicAdd(&cnt[g], 1.0f);
}

// K8: result[g,c] = (pooled[g]/max(cnt,1)) @ Wc + bc  (32x2 output)
__global__ void final_kernel(const float* __restrict__ pooled,
                             const float* __restrict__ cnt,
                             const float* __restrict__ Wc,
                             const float* __restrict__ bc,
                             float* __restrict__ out) {
    int tid = threadIdx.x;
    if (tid >= G_GRAPHS * C_CLS) return;
    int g = tid >> 1, c = tid & 1;
    float inv = 1.0f / fmaxf(cnt[g], 1.0f);
    float sum = bc[c];
    for (int h = 0; h < H_DIM; h++)
        sum += pooled[g * H_DIM + h] * inv * Wc[h * C_CLS + c];
    out[g * C_CLS + c] = sum;
}

// ---------------------------------------------------------------------------
extern "C" void kernel_launch(void* const* d_in, const int* in_sizes, int n_in,
                              void* d_out, int out_size, void* d_ws, size_t ws_size,
                              hipStream_t stream) {
    const float* x         = (const float*)d_in[0];
    const int*   ei        = (const int*)  d_in[1];   // (2, E): row0 src, row1 dst
    const float* ea        = (const float*)d_in[2];
    const int*   batch     = (const int*)  d_in[3];
    const float* Wl        = (const float*)d_in[4];
    const float* bl        = (const float*)d_in[5];
    const float* Wr        = (const float*)d_in[6];
    const float* br        = (const float*)d_in[7];
    const float* We        = (const float*)d_in[8];
    const float* att       = (const float*)d_in[9];
    const float* bias_conv = (const float*)d_in[10];
    const float* Wc        = (const float*)d_in[11];
    const float* bc        = (const float*)d_in[12];
    float* out = (float*)d_out;

    const int* srcp = ei;
    const int* dstp = ei + N_EDGES;

    // Workspace layout (floats), total ~21M floats (~84 MB)
    float* ws     = (float*)d_ws;
    float* deg    = ws;                                   // N
    float* loop   = deg    + N_NODES;                     // N*DE
    float* xl     = loop   + (size_t)N_NODES * DE_DIM;    // N*H
    float* xr     = xl     + (size_t)N_NODES * H_DIM;     // N*H
    float* lg     = xr     + (size_t)N_NODES * H_DIM;     // E+N (logit -> ex)
    float* mx     = lg     + ROWS_TOT;                    // N
    float* den    = mx     + N_NODES;                     // N
    float* outacc = den    + N_NODES;                     // N*H
    float* pooled = outacc + (size_t)N_NODES * H_DIM;     // G*H
    float* cnt    = pooled + G_GRAPHS * H_DIM;            // G

    // init: deg+loop = 0, mx = -inf, den..cnt = 0 (contiguous)
    long long z1 = (long long)N_NODES * (1 + DE_DIM);
    fill_kernel<<<(unsigned)((z1 + 255) / 256), 256, 0, stream>>>(deg, z1, 0.0f);
    fill_kernel<<<(N_NODES + 255) / 256, 256, 0, stream>>>(mx, N_NODES, -INFINITY);
    long long z2 = (long long)N_NODES + (long long)N_NODES * H_DIM
                 + G_GRAPHS * H_DIM + G_GRAPHS;
    fill_kernel<<<(unsigned)((z2 + 255) / 256), 256, 0, stream>>>(den, z2, 0.0f);

    edge_deg_kernel<<<(N_EDGES * DE_DIM + 255) / 256, 256, 0, stream>>>(dstp, ea, deg, loop);
    loop_div_kernel<<<(N_NODES * DE_DIM + 255) / 256, 256, 0, stream>>>(loop, deg);

    gemm_xlxr_kernel<<<N_NODES / 16, 256, 0, stream>>>(x, Wl, bl, Wr, br, xl, xr);

    edge_logit_kernel<<<(ROWS_TOT + 7) / 8, 256, 0, stream>>>(
        srcp, dstp, ea, loop, xl, xr, We, att, lg, mx);
    edge_exp_kernel<<<(ROWS_TOT + 255) / 256, 256, 0, stream>>>(dstp, lg, mx, den);

    long long aggN = (long long)ROWS_TOT * H_DIM;
    edge_agg_kernel<<<(unsigned)((aggN + 255) / 256), 256, 0, stream>>>(
        srcp, dstp, lg, den, xl, outacc);

    node_pool_kernel<<<(N_NODES * H_DIM + 255) / 256, 256, 0, stream>>>(
        outacc, bias_conv, batch, pooled, cnt);
    final_kernel<<<1, 64, 0, stream>>>(pooled, cnt, Wc, bc, out);
}